// Instance_loss_19585050870000
// MI455X (gfx1250) — compile-verified
//
#include <hip/hip_runtime.h>
#include <hip/hip_bf16.h>
#include <math.h>

// Fixed geometry from the reference.
#define B_   8
#define H_   768
#define W_   1024
#define BH_  15
#define BW_  11
#define BOXEL (BH_*BW_)      // 165
#define PH_  (H_/2)          // 384
#define PW_  (W_/2)          // 512

typedef float v2f __attribute__((ext_vector_type(2)));
typedef float v8f __attribute__((ext_vector_type(8)));

// ---------------------------------------------------------------------------
// Exact wave32 sum using the f32 matrix core: A (16x4) x ones(4x16) + 0.
// Lane L<16 supplies A[L][0]=p, A[L][1]=0; lane L>=16 supplies A[L-16][2]=p,
// A[L-16][3]=0  (per the ISA 16x4 f32 A layout).  With B==1 everywhere,
// D[m][n] = p[m] + p[m+16] (identical for all n).  Summing a lane's 8 D VGPRs
// gives rows 0..7 (lanes<16) or rows 8..15 (lanes>=16); one shfl_xor(16)
// completes the exact f32 reduction.
// ---------------------------------------------------------------------------
__device__ __forceinline__ float wave_sum32(float p) {
#if __has_builtin(__builtin_amdgcn_wmma_f32_16x16x4_f32)
    v2f a; a[0] = p;    a[1] = 0.0f;
    v2f b; b[0] = 1.0f; b[1] = 1.0f;
    v8f c = {};
    c = __builtin_amdgcn_wmma_f32_16x16x4_f32(false, a, false, b,
                                              (short)0, c, false, false);
    float s = c[0] + c[1] + c[2] + c[3] + c[4] + c[5] + c[6] + c[7];
    s += __shfl_xor(s, 16, 32);
    return s;
#else
    for (int off = 16; off > 0; off >>= 1) p += __shfl_xor(p, off, 32);
    return p;
#endif
}

// ---------------------------------------------------------------------------
// K0: 15x11 normalized Gaussian (sigma = max(11,15)//2 = 7).  The eps
// threshold in fspecial is a provable no-op here (min raw value ~0.47).
// ---------------------------------------------------------------------------
__global__ void k_gauss(float* __restrict__ gauss) {
    __shared__ float raw[BOXEL];
    __shared__ float inv;
    int t = threadIdx.x;
    if (t < BOXEL) {
        float y = (float)(t / BW_) - 7.0f;
        float x = (float)(t % BW_) - 5.0f;
        raw[t] = expf(-(y * y + x * x) * (1.0f / 98.0f));  // 2*sigma^2 = 98
    }
    __syncthreads();
    if (t == 0) {
        float s = 0.0f;
        for (int i = 0; i < BOXEL; ++i) s += raw[i];
        inv = 1.0f / s;
    }
    __syncthreads();
    if (t < BOXEL) gauss[t] = raw[t] * inv;
}

// K1a: zero the instance_weight scratch map.
__global__ void k_zero(float* __restrict__ p, int n) {
    int i = blockIdx.x * blockDim.x + threadIdx.x;
    if (i < n) p[i] = 0.0f;
}

// K1b: scatter the Gaussian into each (disjoint) box.
__global__ void k_scatter(const int* __restrict__ bsh, const int* __restrict__ bsw,
                          const float* __restrict__ gauss, float* __restrict__ iw,
                          int kPerB, int nScat) {
    int tid = blockIdx.x * blockDim.x + threadIdx.x;
    if (tid >= nScat) return;
    int box = tid / BOXEL;
    int e   = tid - box * BOXEL;
    int i   = e / BW_;
    int j   = e - i * BW_;
    int b   = box / kPerB;
    int sh  = bsh[box];
    int sw  = bsw[box];
    size_t idx = ((size_t)b * H_ + (size_t)(sh + i)) * W_ + (size_t)(sw + j);
    iw[idx] = gauss[e];
}

// K2: 9x9 stride-2 maxpool of gt and instance_weight (window start 2y-4,2x-4,
// clamped; inputs are >= 0 so identity 0 == -inf padding semantics).
__global__ void k_pool(const float* __restrict__ gt, const float* __restrict__ iw,
                       float* __restrict__ pg, float* __restrict__ pi, int nPool) {
    int tid = blockIdx.x * blockDim.x + threadIdx.x;
    if (tid >= nPool) return;
    int b   = tid / (PH_ * PW_);
    int rem = tid - b * (PH_ * PW_);
    int y   = rem / PW_;
    int x   = rem - y * PW_;
    int r0  = 2 * y - 4;
    int c0  = 2 * x - 4;
    const float* gB = gt + (size_t)b * H_ * W_;
    const float* iB = iw + (size_t)b * H_ * W_;
    float mg = 0.0f, mi = 0.0f;
    #pragma unroll
    for (int dr = 0; dr < 9; ++dr) {
        int r = r0 + dr;
        if ((unsigned)r >= (unsigned)H_) continue;
        const float* gR = gB + (size_t)r * W_;
        const float* iR = iB + (size_t)r * W_;
        #pragma unroll
        for (int dc = 0; dc < 9; ++dc) {
            int c = c0 + dc;
            if ((unsigned)c >= (unsigned)W_) continue;
            mg = fmaxf(mg, gR[c]);
            mi = fmaxf(mi, iR[c]);
        }
    }
    pg[tid] = mg;
    pi[tid] = mi;
}

// K3: fused elementwise outputs (dilation_mask, threshold_gt, all_weight+1).
__global__ void k_elem(const float* __restrict__ gt, const float* __restrict__ iw,
                       const float* __restrict__ pg, const float* __restrict__ pi,
                       float* __restrict__ dil, float* __restrict__ thr,
                       float* __restrict__ allw, int nPix) {
    int idx = blockIdx.x * blockDim.x + threadIdx.x;
    if (idx >= nPix) return;
    int b   = idx / (H_ * W_);
    int rem = idx - b * (H_ * W_);
    int h   = rem / W_;
    int w   = rem - h * W_;
    size_t pidx = ((size_t)b * PH_ + (size_t)(h >> 1)) * PW_ + (size_t)(w >> 1);
    float g     = gt[idx];
    float dm    = pg[pidx];
    float outer = dm - g;
    dil[idx]  = dm;
    thr[idx]  = outer * 0.7f + g * 0.2f;
    allw[idx] = pi[pidx] * outer + iw[idx] + 1.0f;
}

// K4: one wave32 per box; lanes sweep the 165 box pixels, then an exact
// matrix-core wave reduction produces one partial per box.
__global__ void k_loss(const float* __restrict__ binar, const float* __restrict__ gt,
                       const int* __restrict__ bsh, const int* __restrict__ bsw,
                       const float* __restrict__ gauss, float* __restrict__ partial,
                       int kPerB, int nBoxes) {
    int wave = threadIdx.x >> 5;
    int lane = threadIdx.x & 31;
    int box  = blockIdx.x * (blockDim.x >> 5) + wave;
    if (box >= nBoxes) return;          // wave-uniform branch: EXEC stays full
    int b  = box / kPerB;
    int sh = bsh[box];
    int sw = bsw[box];
    const float* binB = binar + (size_t)b * H_ * W_;
    const float* gtB  = gt    + (size_t)b * H_ * W_;
    float p = 0.0f;
    for (int e = lane; e < BOXEL; e += 32) {
        int i = e / BW_;
        int j = e - i * BW_;
        size_t idx = (size_t)(sh + i) * W_ + (size_t)(sw + j);
        p += gauss[e] * fabsf(binB[idx] - gtB[idx]);
    }
    float s = wave_sum32(p);
    if (lane == 0) partial[box] = s;
}

// K5: deterministic fixed-order reduction of the per-box partials.
__global__ void k_final(const float* __restrict__ partial, int nBoxes,
                        float* __restrict__ lossOut) {
    __shared__ float sh[256];
    int t = threadIdx.x;
    float s = 0.0f;
    for (int i = t; i < nBoxes; i += 256) s += partial[i];
    sh[t] = s;
    __syncthreads();
    for (int off = 128; off > 0; off >>= 1) {
        if (t < off) sh[t] += sh[t + off];
        __syncthreads();
    }
    if (t == 0) lossOut[0] = sh[0] / (1e-10f + (float)nBoxes);
}

extern "C" void kernel_launch(void* const* d_in, const int* in_sizes, int n_in,
                              void* d_out, int out_size, void* d_ws, size_t ws_size,
                              hipStream_t stream) {
    const float* binar = (const float*)d_in[0];   // [B,1,H,W] f32
    const float* gt    = (const float*)d_in[1];   // [B,1,H,W] f32
    const int*   bsh   = (const int*)d_in[2];     // [B,K] i32
    const int*   bsw   = (const int*)d_in[3];     // [B,K] i32
    float* out = (float*)d_out;                   // loss | dilation | threshold | all_weight+1

    const int nPix   = in_sizes[0];               // B*H*W = 6,291,456
    const int nBoxes = in_sizes[2];               // B*K   = 3,072
    const int kPerB  = nBoxes / B_;               // 384

    // Workspace layout (floats): gauss(256 pad) | iw(nPix) | pooledG | pooledI | partials
    float* ws      = (float*)d_ws;
    float* gauss   = ws;
    float* iw      = ws + 256;
    float* pooledG = iw + (size_t)nPix;
    float* pooledI = pooledG + (size_t)B_ * PH_ * PW_;
    float* partial = pooledI + (size_t)B_ * PH_ * PW_;

    float* lossOut = out;
    float* dil  = out + 1;
    float* thr  = dil + nPix;
    float* allw = thr + nPix;

    const int nScat = nBoxes * BOXEL;             // 506,880
    const int nPool = B_ * PH_ * PW_;             // 1,572,864

    k_gauss  <<<1, 256, 0, stream>>>(gauss);
    k_zero   <<<(nPix  + 255) / 256, 256, 0, stream>>>(iw, nPix);
    k_scatter<<<(nScat + 255) / 256, 256, 0, stream>>>(bsh, bsw, gauss, iw, kPerB, nScat);
    k_pool   <<<(nPool + 255) / 256, 256, 0, stream>>>(gt, iw, pooledG, pooledI, nPool);
    k_elem   <<<(nPix  + 255) / 256, 256, 0, stream>>>(gt, iw, pooledG, pooledI,
                                                       dil, thr, allw, nPix);
    k_loss   <<<(nBoxes + 7) / 8, 256, 0, stream>>>(binar, gt, bsh, bsw, gauss,
                                                    partial, kPerB, nBoxes);
    k_final  <<<1, 256, 0, stream>>>(partial, nBoxes, lossOut);
}